// SageLayer_2826088481577
// MI455X (gfx1250) — compile-verified
//
#include <hip/hip_runtime.h>

// ---------------------------------------------------------------------------
// SageLayer for MI455X (gfx1250, wave32).
// Phase 1 (dominant): random-row gather+mean from 512MB feat table -> HBM bound.
// Phases 2/3: fp32 GEMMs via V_WMMA_F32_16X16X4_F32 (full precision, matrix pipe).
// Phase 4: row L2-normalize + batch gather, wave-per-row.
// ---------------------------------------------------------------------------

typedef float v2f __attribute__((ext_vector_type(2)));
typedef float v8f __attribute__((ext_vector_type(8)));

#define N_UNIQ  20000
#define D_IN    128
#define D_OUT   384
#define K_NBR   32
#define BATCH   40000

// ---------------------------------------------------------------------------
// Kernel A: neighbor mean aggregation + self-row copy.
// One 128-thread block per unique node; thread = feature channel.
// Each neighbor row is 512 contiguous bytes -> coalesced global_load_b32.
// ---------------------------------------------------------------------------
__global__ __launch_bounds__(128)
void agg_kernel(const int* __restrict__ unique_nodes,
                const int* __restrict__ adj,
                const int* __restrict__ dis,
                const float* __restrict__ feat,
                float* __restrict__ Xself,
                float* __restrict__ Xadj,
                float* __restrict__ Xdis) {
    const int n  = blockIdx.x;
    const int ch = threadIdx.x;          // 0..127

    const int sn = unique_nodes[n];
    Xself[n * D_IN + ch] = feat[(long long)sn * D_IN + ch];

    float sa = 0.f, sd = 0.f;
#pragma unroll 4
    for (int k = 0; k < K_NBR; ++k) {
        const int na = adj[n * K_NBR + k];   // wave-uniform -> scalar load
        const int nd = dis[n * K_NBR + k];
        sa += feat[(long long)na * D_IN + ch];
        sd += feat[(long long)nd * D_IN + ch];
    }
    Xadj[n * D_IN + ch] = sa * (1.0f / K_NBR);
    Xdis[n * D_IN + ch] = sd * (1.0f / K_NBR);
}

// ---------------------------------------------------------------------------
// Wave-level 16x16 fp32 GEMM tile: D = A[row0:+16, 0:K] * B[0:K, col0:+16]
// using V_WMMA_F32_16X16X4_F32, K stepped by 4.
// A 16x4 layout: lanes 0-15 {VGPR0=K+0, VGPR1=K+1}, lanes 16-31 {K+2, K+3}.
// B 4x16 layout mirrors A. C/D: VGPR v -> M=v (lanes 0-15), M=v+8 (16-31).
// Two accumulators interleaved to relax the WMMA->WMMA RAW hazard.
// ---------------------------------------------------------------------------
__device__ __forceinline__
v8f wmma_tile_f32(const float* __restrict__ A, int lda,
                  const float* __restrict__ B, int ldb,
                  int Kdim, int row0, int col0, int lane) {
    v8f acc0 = {};
    v8f acc1 = {};
    const int lhalf = lane & 15;
    const int khi   = (lane >> 4) << 1;           // 0 (lanes 0-15) or 2 (16-31)
    const float* arow = A + (long long)(row0 + lhalf) * lda + khi;
    const float* bcol = B + (long long)khi * ldb + col0 + lhalf;

    for (int k = 0; k < Kdim; k += 8) {
        v2f a, b;
        a.x = arow[k];
        a.y = arow[k + 1];
        b.x = bcol[(long long)k * ldb];
        b.y = bcol[(long long)(k + 1) * ldb];
        acc0 = __builtin_amdgcn_wmma_f32_16x16x4_f32(
            false, a, false, b, (short)0, acc0, false, false);

        a.x = arow[k + 4];
        a.y = arow[k + 5];
        b.x = bcol[(long long)(k + 4) * ldb];
        b.y = bcol[(long long)(k + 5) * ldb];
        acc1 = __builtin_amdgcn_wmma_f32_16x16x4_f32(
            false, a, false, b, (short)0, acc1, false, false);
    }
    return acc0 + acc1;
}

// ---------------------------------------------------------------------------
// Kernel B: H1[N,384] = [Xself@W_self | Xadj@W_adj | Xdis@W_dis] + bias.
// Tiles: 1250 row-tiles x 24 col-tiles; col-tile/8 selects the weight block.
// 8 waves per 256-thread block, one tile per wave; 30000 tiles = 3750 blocks.
// ---------------------------------------------------------------------------
__global__ __launch_bounds__(256)
void gemm1_kernel(const float* __restrict__ Xself,
                  const float* __restrict__ Xadj,
                  const float* __restrict__ Xdis,
                  const float* __restrict__ Wself,
                  const float* __restrict__ Wadj,
                  const float* __restrict__ Wdis,
                  const float* __restrict__ bias,
                  float* __restrict__ H1) {
    const int wave = threadIdx.x >> 5;
    const int lane = threadIdx.x & 31;
    const int tile = blockIdx.x * 8 + wave;
    const int colTiles = D_OUT / 16;              // 24
    if (tile >= (N_UNIQ / 16) * colTiles) return; // wave-uniform guard

    const int rt  = tile / colTiles;
    const int ct  = tile % colTiles;
    const int sel = ct >> 3;                      // 0=self, 1=adj, 2=dis
    const int ctl = ct & 7;                       // col tile within 128-wide W

    const float* X = (sel == 0) ? Xself : (sel == 1) ? Xadj : Xdis;
    const float* W = (sel == 0) ? Wself : (sel == 1) ? Wadj : Wdis;

    v8f acc = wmma_tile_f32(X, D_IN, W, D_IN, D_IN, rt * 16, ctl * 16, lane);

    const int col = ct * 16 + (lane & 15);        // global output column
    const float bv = bias[col];
    const int m0 = rt * 16 + ((lane >> 4) << 3);  // +8 rows for upper lanes
#pragma unroll
    for (int v = 0; v < 8; ++v)
        H1[(long long)(m0 + v) * D_OUT + col] = acc[v] + bv;
}

// ---------------------------------------------------------------------------
// Kernel C: F = leakyrelu(H1 @ WC_w + WC_b, 0.2).  K=384, same tiling.
// ---------------------------------------------------------------------------
__global__ __launch_bounds__(256)
void gemm2_kernel(const float* __restrict__ H1,
                  const float* __restrict__ WC,
                  const float* __restrict__ WCb,
                  float* __restrict__ F) {
    const int wave = threadIdx.x >> 5;
    const int lane = threadIdx.x & 31;
    const int tile = blockIdx.x * 8 + wave;
    const int colTiles = D_OUT / 16;              // 24
    if (tile >= (N_UNIQ / 16) * colTiles) return;

    const int rt = tile / colTiles;
    const int ct = tile % colTiles;

    v8f acc = wmma_tile_f32(H1, D_OUT, WC, D_OUT, D_OUT, rt * 16, ct * 16, lane);

    const int col = ct * 16 + (lane & 15);
    const float bv = WCb[col];
    const int m0 = rt * 16 + ((lane >> 4) << 3);
#pragma unroll
    for (int v = 0; v < 8; ++v) {
        float x = acc[v] + bv;
        x = (x >= 0.f) ? x : 0.2f * x;            // LeakyReLU(0.2)
        F[(long long)(m0 + v) * D_OUT + col] = x;
    }
}

// ---------------------------------------------------------------------------
// Kernel D: out[b] = F[nodes_idx[b]] / max(||.||2, 1e-12).
// One wave32 per output row: 12 floats/lane, shfl_xor tree reduction.
// ---------------------------------------------------------------------------
__global__ __launch_bounds__(256)
void norm_gather_kernel(const int* __restrict__ nodes_idx,
                        const float* __restrict__ F,
                        float* __restrict__ out) {
    const int wave = threadIdx.x >> 5;
    const int lane = threadIdx.x & 31;
    const int b = blockIdx.x * 8 + wave;
    if (b >= BATCH) return;

    const int n = nodes_idx[b];
    const float* row = F + (long long)n * D_OUT;

    float vals[D_OUT / 32];
    float ss = 0.f;
#pragma unroll
    for (int i = 0; i < D_OUT / 32; ++i) {
        vals[i] = row[lane + i * 32];
        ss += vals[i] * vals[i];
    }
#pragma unroll
    for (int off = 16; off > 0; off >>= 1)
        ss += __shfl_xor(ss, off, 32);

    const float inv = 1.0f / fmaxf(sqrtf(ss), 1e-12f);
    float* o = out + (long long)b * D_OUT;
#pragma unroll
    for (int i = 0; i < D_OUT / 32; ++i)
        o[lane + i * 32] = vals[i] * inv;
}

// ---------------------------------------------------------------------------
extern "C" void kernel_launch(void* const* d_in, const int* in_sizes, int n_in,
                              void* d_out, int out_size, void* d_ws, size_t ws_size,
                              hipStream_t stream) {
    const int*   unique_nodes = (const int*)d_in[0];
    const int*   adj          = (const int*)d_in[1];
    const int*   dis          = (const int*)d_in[2];
    const int*   nodes_idx    = (const int*)d_in[3];
    const float* feat         = (const float*)d_in[4];
    const float* W_self       = (const float*)d_in[5];
    const float* W_adj        = (const float*)d_in[6];
    const float* W_dis        = (const float*)d_in[7];
    const float* bias         = (const float*)d_in[8];
    const float* WC_w         = (const float*)d_in[9];
    const float* WC_b         = (const float*)d_in[10];
    float* out = (float*)d_out;

    // Workspace layout (floats): Xself|Xadj|Xdis|H1|F  ~= 92 MB total.
    float* Xself = (float*)d_ws;
    float* Xadj  = Xself + (size_t)N_UNIQ * D_IN;
    float* Xdis  = Xadj  + (size_t)N_UNIQ * D_IN;
    float* H1    = Xdis  + (size_t)N_UNIQ * D_IN;
    float* F     = H1    + (size_t)N_UNIQ * D_OUT;

    // Phase 1: HBM-bound gather+mean (~660 MB @ 23.3 TB/s).
    agg_kernel<<<N_UNIQ, 128, 0, stream>>>(unique_nodes, adj, dis, feat,
                                           Xself, Xadj, Xdis);

    // Phase 2: three fused 20000x128x128 fp32 WMMA GEMMs + bias.
    const int tiles1 = (N_UNIQ / 16) * (D_OUT / 16);       // 30000
    gemm1_kernel<<<tiles1 / 8, 256, 0, stream>>>(Xself, Xadj, Xdis,
                                                 W_self, W_adj, W_dis,
                                                 bias, H1);

    // Phase 3: 20000x384x384 fp32 WMMA GEMM + bias + LeakyReLU.
    gemm2_kernel<<<tiles1 / 8, 256, 0, stream>>>(H1, WC_w, WC_b, F);

    // Phase 4: L2-normalize + batch gather (wave per output row).
    norm_gather_kernel<<<BATCH / 8, 256, 0, stream>>>(nodes_idx, F, out);
}